// Net_gcn_59768764892009
// MI455X (gfx1250) — compile-verified
//
#include <hip/hip_runtime.h>
#include <cstdint>
#include <cstddef>

#define N_NODES   100000
#define N_EDGES   3200000
#define D_FEAT    512
#define HIDDEN    16
#define N_CLASSES 6
#define N_TILES   (N_NODES / 16)   // 6250 (exact)
#define WPB1      4                // waves per block for GEMM1 (128 threads)
#define WPB2      8                // waves per block for GEMM2 (256 threads)
#define KC        32               // K-chunk staged in LDS
#define AS        36               // padded LDS row stride in floats (bank = 4M+c mod 64, conflict-free)

typedef float v2f __attribute__((ext_vector_type(2)));
typedef float v8f __attribute__((ext_vector_type(8)));
typedef int   v4i __attribute__((ext_vector_type(4)));

// Async direct-to-LDS path (gfx1250): guarded so both toolchain variants and the
// host pass still compile; fallback is the proven sync global->VGPR->LDS path.
#if defined(__AMDGCN__) && __has_builtin(__builtin_amdgcn_global_load_async_to_lds_b128) && \
    __has_builtin(__builtin_amdgcn_s_wait_asynccnt)
#define ASYNC_LDS 1
#else
#define ASYNC_LDS 0
#endif

// ---------------- degree / normalization ----------------

__global__ __launch_bounds__(256) void k_init_deg(float* __restrict__ deg) {
    int i = blockIdx.x * 256 + threadIdx.x;
    if (i < N_NODES) deg[i] = 1.0f;  // self-loop weight
}

__global__ __launch_bounds__(256) void k_deg_scatter(const int* __restrict__ ei,
                                                     const float* __restrict__ ew,
                                                     float* __restrict__ deg) {
    int e = blockIdx.x * 256 + threadIdx.x;
    if (e < N_EDGES) atomicAdd(&deg[ei[N_EDGES + e]], ew[e]);
}

__global__ __launch_bounds__(256) void k_dinv(float* __restrict__ deg) {
    int i = blockIdx.x * 256 + threadIdx.x;
    if (i < N_NODES) { float d = deg[i]; deg[i] = d > 0.0f ? rsqrtf(d) : 0.0f; }
}

// ---------------- GEMM1: h1 = x @ W1 via v_wmma_f32_16x16x4_f32 ----------------
// One wave per 16-row tile; W1 pre-swizzled in LDS (8B K-pairs per lane);
// A tile double-buffered in wave-private LDS, staged with async-to-LDS loads.

__global__ __launch_bounds__(128) void k_gemm1(const float* __restrict__ x,
                                               const float* __restrict__ W1,
                                               float* __restrict__ h1) {
    __shared__ float lw[(D_FEAT / 2) * 32];     // swizzled W1: pair p holds K=2p,2p+1 adjacent per N
    __shared__ float la[WPB1][2][16 * AS];      // per-wave double-buffered A tile, padded rows

    const int tid  = threadIdx.x;
    const int lane = tid & 31;
    const int wv   = tid >> 5;
    const int tIdx = blockIdx.x * WPB1 + wv;
    const int tile = (tIdx < N_TILES) ? tIdx : (N_TILES - 1);  // clamp: keep EXEC full for WMMA
    const int row0 = tile * 16;
    const int m    = lane & 15;   // M for A / N for B / N for D
    const int hf   = lane >> 4;   // lane half selects K offset

    // Preload W1 into LDS, swizzled so each lane's (K, K+1) pair is one aligned 8B read.
    for (int idx = tid; idx < D_FEAT * HIDDEN; idx += 128) {
        int k = idx >> 4, n = idx & 15;
        lw[(k >> 1) * 32 + n * 2 + (k & 1)] = W1[idx];
    }
    __syncthreads();

    const int NCH = D_FEAT / KC;   // 16 chunks
    v8f c = {};

#if ASYNC_LDS
    // ---- async double-buffered pipeline: no barriers in the k-loop ----
    auto stage = [&](int kc, int buf) {
#pragma unroll
        for (int i = 0; i < 4; ++i) {           // 4 x (32 lanes x 16B) = 2048B chunk
            int s  = i * 32 + lane;             // float4 slot 0..127
            int r  = s >> 3;                    // row 0..15 (8 float4 per 32-float row)
            int cb = (s & 7) * 4;               // float offset in row
            const float* g = x + (size_t)(row0 + r) * D_FEAT + kc + cb;
            float*       l = &la[wv][buf][r * AS + cb];
            __builtin_amdgcn_global_load_async_to_lds_b128(
                (__attribute__((address_space(1))) v4i*)g,
                (__attribute__((address_space(3))) v4i*)l, 0, 0);
        }
    };

    stage(0, 0);
    for (int ci = 0; ci < NCH; ++ci) {
        if (ci + 1 < NCH) {
            stage((ci + 1) * KC, (ci + 1) & 1);
            __builtin_amdgcn_s_wait_asynccnt(4);  // in-order: chunk ci has landed
        } else {
            __builtin_amdgcn_s_wait_asynccnt(0);
        }
        asm volatile("" ::: "memory");
        const int kc  = ci * KC;
        const int buf = ci & 1;
#pragma unroll
        for (int kk = 0; kk < KC; kk += 4) {
            v2f a = *(const v2f*)(&la[wv][buf][m * AS + kk + 2 * hf]);
            v2f b = *(const v2f*)(&lw[(((kc + kk) >> 1) + hf) * 32 + m * 2]);
            c = __builtin_amdgcn_wmma_f32_16x16x4_f32(false, a, false, b,
                                                      (short)0, c, false, false);
        }
    }
#else
    // ---- sync fallback: stage via VGPRs with block barriers ----
    for (int ci = 0; ci < NCH; ++ci) {
        const int kc  = ci * KC;
        const int buf = ci & 1;
#pragma unroll
        for (int i = 0; i < 4; ++i) {
            int s  = i * 32 + lane;
            int r  = s >> 3;
            int cb = (s & 7) * 4;
            const float4 v = *(const float4*)(x + (size_t)(row0 + r) * D_FEAT + kc + cb);
            *(float4*)(&la[wv][buf][r * AS + cb]) = v;
        }
        __syncthreads();
#pragma unroll
        for (int kk = 0; kk < KC; kk += 4) {
            v2f a = *(const v2f*)(&la[wv][buf][m * AS + kk + 2 * hf]);
            v2f b = *(const v2f*)(&lw[(((kc + kk) >> 1) + hf) * 32 + m * 2]);
            c = __builtin_amdgcn_wmma_f32_16x16x4_f32(false, a, false, b,
                                                      (short)0, c, false, false);
        }
        __syncthreads();
    }
#endif

    if (tIdx < N_TILES) {
        for (int r = 0; r < 8; ++r)  // D: vgpr r -> row (r + 8*hf), col m
            h1[(size_t)(row0 + r + 8 * hf) * HIDDEN + m] = c[r];
    }
}

// ---------------- fused bias + self-loop: x1 = b1 + dinv^2 * h1 ----------------

__global__ __launch_bounds__(256) void k_fuse1(const float* __restrict__ h1,
                                               const float* __restrict__ dinv,
                                               const float* __restrict__ b1,
                                               float* __restrict__ x1) {
    int i = blockIdx.x * 256 + threadIdx.x;
    if (i < N_NODES * HIDDEN) {
        int nd = i >> 4;
        float di = dinv[nd];
        x1[i] = b1[i & 15] + di * di * h1[i];
    }
}

// ---------------- layer-1 edge scatter: lane per (edge, feature) ----------------

__global__ __launch_bounds__(256) void k_scatter1(const int* __restrict__ ei,
                                                  const float* __restrict__ ew,
                                                  const float* __restrict__ dinv,
                                                  const float* __restrict__ h1,
                                                  float* __restrict__ x1) {
    int t = blockIdx.x * 256 + threadIdx.x;   // E*16 = 51.2M fits int32
    int e = t >> 4;
    int j = t & 15;
    if (e < N_EDGES) {
        int r  = ei[e];
        int cn = ei[N_EDGES + e];
        float w = dinv[r] * ew[e] * dinv[cn];
        atomicAdd(&x1[(size_t)cn * HIDDEN + j], w * h1[(size_t)r * HIDDEN + j]);
    }
}

// ---------------- GEMM2: h2 = x1 @ W2 via WMMA, N padded to 16 ----------------

__global__ __launch_bounds__(256) void k_gemm2(const float* __restrict__ x1,
                                               const float* __restrict__ W2,
                                               float* __restrict__ h2) {
    const int tid  = threadIdx.x;
    const int lane = tid & 31;
    const int wv   = tid >> 5;
    const int tIdx = blockIdx.x * WPB2 + wv;
    const int tile = (tIdx < N_TILES) ? tIdx : (N_TILES - 1);
    const int row0 = tile * 16;
    const int n    = lane & 15;
    const int hf   = lane >> 4;

    v8f c = {};
    for (int k = 0; k < HIDDEN; k += 4) {
        int ka = k + 2 * hf;
        v2f a = *(const v2f*)(x1 + (size_t)(row0 + n) * HIDDEN + ka);  // 8B-aligned (ka even)
        v2f b;
        b.x = (n < N_CLASSES) ? W2[ka * N_CLASSES + n]       : 0.0f;   // zero-pad N -> 16
        b.y = (n < N_CLASSES) ? W2[(ka + 1) * N_CLASSES + n] : 0.0f;
        c = __builtin_amdgcn_wmma_f32_16x16x4_f32(false, a, false, b,
                                                  (short)0, c, false, false);
    }
    if (tIdx < N_TILES && n < N_CLASSES) {
        for (int r = 0; r < 8; ++r)
            h2[(size_t)(row0 + r + 8 * hf) * N_CLASSES + n] = c[r];
    }
}

// ---------------- fused bias + self-loop layer 2 ----------------

__global__ __launch_bounds__(256) void k_fuse2(const float* __restrict__ h2,
                                               const float* __restrict__ dinv,
                                               const float* __restrict__ b2,
                                               float* __restrict__ x2) {
    int i = blockIdx.x * 256 + threadIdx.x;
    if (i < N_NODES * N_CLASSES) {
        int nd = i / N_CLASSES;
        int j  = i - nd * N_CLASSES;
        float di = dinv[nd];
        x2[i] = b2[j] + di * di * h2[i];
    }
}

// ---------------- layer-2 edge scatter ----------------

__global__ __launch_bounds__(256) void k_scatter2(const int* __restrict__ ei,
                                                  const float* __restrict__ ew,
                                                  const float* __restrict__ dinv,
                                                  const float* __restrict__ h2,
                                                  float* __restrict__ x2) {
    int t = blockIdx.x * 256 + threadIdx.x;   // E*8 = 25.6M
    int e = t >> 3;
    int j = t & 7;
    if (e < N_EDGES && j < N_CLASSES) {
        int r  = ei[e];
        int cn = ei[N_EDGES + e];
        float w = dinv[r] * ew[e] * dinv[cn];
        atomicAdd(&x2[(size_t)cn * N_CLASSES + j], w * h2[(size_t)r * N_CLASSES + j]);
    }
}

// ---------------- in-place log_softmax over 6 classes ----------------

__global__ __launch_bounds__(256) void k_logsoftmax(float* __restrict__ z) {
    int i = blockIdx.x * 256 + threadIdx.x;
    if (i < N_NODES) {
        float v[N_CLASSES];
        float mx = -3.402823466e+38f;
        for (int j = 0; j < N_CLASSES; ++j) {
            v[j] = z[(size_t)i * N_CLASSES + j];
            mx = fmaxf(mx, v[j]);
        }
        float s = 0.0f;
        for (int j = 0; j < N_CLASSES; ++j) s += expf(v[j] - mx);
        float lse = mx + logf(s);
        for (int j = 0; j < N_CLASSES; ++j) z[(size_t)i * N_CLASSES + j] = v[j] - lse;
    }
}

// ---------------- launch ----------------

extern "C" void kernel_launch(void* const* d_in, const int* in_sizes, int n_in,
                              void* d_out, int out_size, void* d_ws, size_t ws_size,
                              hipStream_t stream) {
    const float* x  = (const float*)d_in[0];
    const float* W1 = (const float*)d_in[1];
    const float* b1 = (const float*)d_in[2];
    const float* W2 = (const float*)d_in[3];
    const float* b2 = (const float*)d_in[4];
    const float* ew = (const float*)d_in[5];
    const int*   ei = (const int*)d_in[6];

    float* out  = (float*)d_out;
    float* logp = out;                                   // [N, 6]  (output 0)
    float* x1   = out + (size_t)N_NODES * N_CLASSES;     // [N, 16] (output 1) == x1

    float* ws  = (float*)d_ws;
    float* deg = ws;                                     // N floats (becomes dinv in place)
    float* h1  = ws + N_NODES;                           // N*16
    float* h2  = h1 + (size_t)N_NODES * HIDDEN;          // N*6

    k_init_deg   <<<(N_NODES + 255) / 256, 256, 0, stream>>>(deg);
    k_deg_scatter<<<(N_EDGES + 255) / 256, 256, 0, stream>>>(ei, ew, deg);
    k_dinv       <<<(N_NODES + 255) / 256, 256, 0, stream>>>(deg);

    k_gemm1   <<<(N_TILES + WPB1 - 1) / WPB1, 128, 0, stream>>>(x, W1, h1);
    k_fuse1   <<<(N_NODES * HIDDEN + 255) / 256, 256, 0, stream>>>(h1, deg, b1, x1);
    k_scatter1<<<(N_EDGES * HIDDEN) / 256, 256, 0, stream>>>(ei, ew, deg, h1, x1);

    k_gemm2   <<<(N_TILES + WPB2 - 1) / WPB2, 256, 0, stream>>>(x1, W2, h2);
    k_fuse2   <<<(N_NODES * N_CLASSES + 255) / 256, 256, 0, stream>>>(h2, deg, b2, logp);
    k_scatter2<<<(N_EDGES * 8) / 256, 256, 0, stream>>>(ei, ew, deg, h2, logp);

    k_logsoftmax<<<(N_NODES + 255) / 256, 256, 0, stream>>>(logp);
}